// L2ContrastiveLoss_15496242003988
// MI455X (gfx1250) — compile-verified
//
#include <hip/hip_runtime.h>

typedef float v2f __attribute__((ext_vector_type(2)));
typedef float v8f __attribute__((ext_vector_type(8)));

#define N_ROWS 8192
#define DIMS 128
#define TILE 128          // block tile: 128 rows of f1 x 128 rows of f2
#define LDS_STRIDE 132    // padded row stride (floats): 132%64=4 -> conflict-free, 528B keeps 16B align
#define MARGIN 1.0f

// ---------------- pass 0: row squared norms (one wave per row) ----------------
__global__ void row_sqnorm_kernel(const float* __restrict__ f1,
                                  const float* __restrict__ f2,
                                  float* __restrict__ sq) {
  int gwave = (blockIdx.x * blockDim.x + threadIdx.x) >> 5;
  int lane  = threadIdx.x & 31;
  if (gwave >= 2 * N_ROWS) return;
  const float* src = (gwave < N_ROWS) ? (f1 + (size_t)gwave * DIMS)
                                      : (f2 + (size_t)(gwave - N_ROWS) * DIMS);
  float4 v = ((const float4*)src)[lane];          // 32 lanes x 4 = 128 elems
  float acc = v.x * v.x + v.y * v.y + v.z * v.z + v.w * v.w;
  #pragma unroll
  for (int off = 16; off >= 1; off >>= 1) acc += __shfl_xor(acc, off, 32);
  if (lane == 0) sq[gwave] = acc;
}

// ---------------- pass 1: tiled WMMA distance + fused loss epilogue ----------------
__global__ void dist_loss_kernel(const float* __restrict__ f1,
                                 const float* __restrict__ f2,
                                 const float* __restrict__ sq1,
                                 const float* __restrict__ sq2,
                                 float* __restrict__ partials) {
  extern __shared__ float smem[];                 // [2][TILE][LDS_STRIDE]
  float* As = smem;                               // f1 tile
  float* Bs = smem + TILE * LDS_STRIDE;           // f2 tile

  const int tid  = threadIdx.x;
  const int wave = tid >> 5;                      // 0..7 : column subtile owner
  const int lane = tid & 31;
  const int hl   = lane >> 4;                     // lane half (0/1)
  const int idx  = lane & 15;                     // 0..15

  const int rowBase = blockIdx.y * TILE;          // rows from feature1
  const int colBase = blockIdx.x * TILE;          // rows from feature2 (= output cols)

  // Stage both 128x128 tiles into LDS (coalesced float4).
  for (int i = tid; i < TILE * (DIMS / 4); i += 256) {
    int r = i >> 5;                               // row 0..127
    int c = (i & 31) << 2;                        // col 0,4,...,124
    float4 a = *(const float4*)(f1 + (size_t)(rowBase + r) * DIMS + c);
    float4 b = *(const float4*)(f2 + (size_t)(colBase + r) * DIMS + c);
    *(float4*)(As + r * LDS_STRIDE + c) = a;
    *(float4*)(Bs + r * LDS_STRIDE + c) = b;
  }
  __syncthreads();

  v8f acc[8];
  #pragma unroll
  for (int r = 0; r < 8; ++r) acc[r] = (v8f){0.f, 0.f, 0.f, 0.f, 0.f, 0.f, 0.f, 0.f};

  // Wave `wave` computes the 128x16 slab: subtiles (r, wave), r = 0..7.
  // f32 WMMA A 16x4 layout: lanes 0-15 hold K=0,1; lanes 16-31 hold K=2,3.
  // B 4x16 = transposed f2 tile: same per-lane float2 indexing from row-major Bs.
  const float* aPtr = As + idx * LDS_STRIDE + 2 * hl;
  const float* bPtr = Bs + (wave * 16 + idx) * LDS_STRIDE + 2 * hl;

  // Software-pipelined K loop: prefetch fragments for k+4 into registers
  // before issuing the 8 WMMAs for k, so DS latency overlaps matrix math.
  v2f aCur[8], aNxt[8], bCur, bNxt;
  bCur = *(const v2f*)(bPtr);
  #pragma unroll
  for (int r = 0; r < 8; ++r) {
    aCur[r] = *(const v2f*)(aPtr + r * 16 * LDS_STRIDE);
    aNxt[r] = aCur[r];
  }
  bNxt = bCur;

  #pragma unroll 4
  for (int k = 4; k <= DIMS; k += 4) {
    if (k < DIMS) {
      bNxt = *(const v2f*)(bPtr + k);
      #pragma unroll
      for (int r = 0; r < 8; ++r)
        aNxt[r] = *(const v2f*)(aPtr + r * 16 * LDS_STRIDE + k);
    }
    #pragma unroll
    for (int r = 0; r < 8; ++r) {
      acc[r] = __builtin_amdgcn_wmma_f32_16x16x4_f32(
          /*neg_a=*/false, aCur[r], /*neg_b=*/false, bCur,
          /*c_mod=*/(short)0, acc[r], /*reuse_a=*/false, /*reuse_b=*/false);
    }
    #pragma unroll
    for (int r = 0; r < 8; ++r) aCur[r] = aNxt[r];
    bCur = bNxt;
  }

  // Stage the row-tile norms into LDS for broadcast reads in the epilogue.
  __syncthreads();                                // all waves done reading tiles
  if (tid < TILE) smem[tid] = sq1[rowBase + tid];
  __syncthreads();

  // Epilogue: d2 = ||a||^2 + ||b||^2 - 2*dot ; diag -> d2 ; off-diag -> relu(1-d)^2
  float part = 0.0f;
  const int gN = colBase + wave * 16 + idx;       // constant per lane
  const float sN = sq2[gN];
  #pragma unroll
  for (int r = 0; r < 8; ++r) {
    #pragma unroll
    for (int e = 0; e < 8; ++e) {
      int rowLocal = r * 16 + e + 8 * hl;         // f32 C/D layout: VGPR e -> M=e(+8)
      int gM = rowBase + rowLocal;
      float d2 = fmaxf(smem[rowLocal] + sN - 2.0f * acc[r][e], 0.0f);
      if (gM == gN) {
        part += d2;                               // diag: d^2 directly
      } else {
        float t = fmaxf(MARGIN - sqrtf(d2), 0.0f);
        part += t * t;
      }
    }
  }

  #pragma unroll
  for (int off = 16; off >= 1; off >>= 1) part += __shfl_xor(part, off, 32);

  if (lane == 0) smem[TILE + wave] = part;        // disjoint from sq1 slots, barrier below
  __syncthreads();
  if (tid == 0) {
    float s = 0.0f;
    #pragma unroll
    for (int w = 0; w < 8; ++w) s += smem[TILE + w];
    partials[blockIdx.y * gridDim.x + blockIdx.x] = s;
  }
}

// ---------------- pass 2: deterministic final reduction ----------------
__global__ void final_reduce_kernel(const float* __restrict__ partials, int n,
                                    float* __restrict__ out) {
  __shared__ float red[256];
  float s = 0.0f;
  for (int i = threadIdx.x; i < n; i += 256) s += partials[i];
  red[threadIdx.x] = s;
  __syncthreads();
  for (int stride = 128; stride > 0; stride >>= 1) {
    if (threadIdx.x < stride) red[threadIdx.x] += red[threadIdx.x + stride];
    __syncthreads();
  }
  if (threadIdx.x == 0) out[0] = red[0] / (2.0f * (float)N_ROWS);
}

extern "C" void kernel_launch(void* const* d_in, const int* in_sizes, int n_in,
                              void* d_out, int out_size, void* d_ws, size_t ws_size,
                              hipStream_t stream) {
  const float* f1 = (const float*)d_in[0];
  const float* f2 = (const float*)d_in[1];
  float* ws       = (float*)d_ws;
  float* sqbuf    = ws;                   // [0..8191]=||f1 rows||^2, [8192..16383]=||f2 rows||^2
  float* sq1      = ws;
  float* sq2      = ws + N_ROWS;
  float* partials = ws + 2 * N_ROWS;      // 64*64 = 4096 block partials

  // Pass 0: 2*8192 rows, 8 waves per 256-thread block.
  row_sqnorm_kernel<<<(2 * N_ROWS) / 8, 256, 0, stream>>>(f1, f2, sqbuf);

  // Pass 1: 64x64 grid of 128x128 output tiles, 132 KB dynamic LDS per block.
  const size_t ldsBytes = (size_t)2 * TILE * LDS_STRIDE * sizeof(float);
  (void)hipFuncSetAttribute((const void*)dist_loss_kernel,
                            hipFuncAttributeMaxDynamicSharedMemorySize,
                            (int)ldsBytes);
  dim3 grid(N_ROWS / TILE, N_ROWS / TILE);
  dist_loss_kernel<<<grid, 256, ldsBytes, stream>>>(f1, f2, sq1, sq2, partials);

  // Pass 2: single-block deterministic sum + scale.
  final_reduce_kernel<<<1, 256, 0, stream>>>(partials,
                                             (N_ROWS / TILE) * (N_ROWS / TILE),
                                             (float*)d_out);
}